// AsCOOT_3968549782209
// MI455X (gfx1250) — compile-verified
//
#include <hip/hip_runtime.h>
#include <math.h>

// COOT (co-optimal transport) for N=M=4096, D1=D2=128, f32.
// d_out = [pi_s (4096*4096) | pi_f (128*128)]  (flat, return order).
// The big kernel matrix K / C_s / pi_s all share the d_out pi_s region
// (in-place transforms), so d_ws only needs ~4.4 MB.
//
// GEMMs use V_WMMA_F32_16X16X4_F32 (f32 needed: eps=0.01 amplifies
// cost-matrix error x100 inside exp). Sinkhorn matvecs stream the
// L2-resident 64MB K with coalesced deterministic reductions.

typedef float v2f __attribute__((ext_vector_type(2)));
typedef float v8f __attribute__((ext_vector_type(8)));

#define N_S   4096
#define M_S   4096
#define D_F   128
#define NTOT  (N_S * M_S)
#define EPSI  0.01f
#define MAX_INNER 100
#define MAX_OUTER 5

// ---------------------------------------------------------------------------
// Generic f32 WMMA GEMM: C[i,j] = alpha * sum_k A(i,k)*B(k,j)  (+ trow[i]+tcol[j])
// One wave computes one 16x16 tile; 8 waves per block; grid sized exactly.
// A(i,k) = TRANS_A ? A[k*lda+i] : A[i*lda+k];  B(k,j) = TRANS_B ? B[j*ldb+k] : B[k*ldb+j]
// ---------------------------------------------------------------------------
template<int TRANS_A, int TRANS_B, int EPI>
__global__ void __launch_bounds__(256)
wmma_gemm_f32(const float* __restrict__ A, int lda,
              const float* __restrict__ B, int ldb,
              float* __restrict__ C, int ldc,
              int N, int K, float alpha,
              const float* __restrict__ trow,
              const float* __restrict__ tcol)
{
    const int lane   = threadIdx.x & 31;
    const int wave   = threadIdx.x >> 5;
    const int tilesN = N >> 4;
    const int tile   = blockIdx.x * 8 + wave;
    const int tm     = tile / tilesN;
    const int tn     = tile - tm * tilesN;
    const int i0     = tm << 4;
    const int j0     = tn << 4;
    const int half   = lane >> 4;   // 0: K pair {0,1}, 1: K pair {2,3}
    const int l      = lane & 15;
    const int ra     = i0 + l;      // A row handled by this lane
    const int cb     = j0 + l;      // B col handled by this lane

    v8f acc;
#pragma unroll
    for (int q = 0; q < 8; ++q) acc[q] = 0.0f;

    for (int kk = 0; kk < K; kk += 4) {
        const int ka = kk + (half << 1);
        v2f av, bv;
        if (TRANS_A) {
            av.x = A[(size_t)(ka    ) * lda + ra];
            av.y = A[(size_t)(ka + 1) * lda + ra];
        } else {
            av.x = A[(size_t)ra * lda + ka];
            av.y = A[(size_t)ra * lda + ka + 1];
        }
        if (TRANS_B) {
            bv.x = B[(size_t)cb * ldb + ka];
            bv.y = B[(size_t)cb * ldb + ka + 1];
        } else {
            bv.x = B[(size_t)(ka    ) * ldb + cb];
            bv.y = B[(size_t)(ka + 1) * ldb + cb];
        }
        // (neg_a, A, neg_b, B, c_mod, C, reuse_a, reuse_b)
        acc = __builtin_amdgcn_wmma_f32_16x16x4_f32(false, av, false, bv,
                                                    (short)0, acc, false, false);
    }

#pragma unroll
    for (int q = 0; q < 8; ++q) {
        const int r = i0 + q + (half << 3);     // C layout: vgpr q -> M=q / M=8+q
        const int c = j0 + l;
        float val = alpha * acc[q];
        if (EPI) val += trow[r] + tcol[c];
        C[(size_t)r * ldc + c] = val;
    }
}

// ---------------------------------------------------------------------------
// Init: pi_f uniform, its row/col sums, pi_f.sum()=1
// ---------------------------------------------------------------------------
__global__ void k_init(float* __restrict__ pif, float* __restrict__ rf,
                       float* __restrict__ cf, float* __restrict__ sc)
{
    int idx = blockIdx.x * 256 + threadIdx.x;
    if (idx < D_F * D_F) pif[idx] = 1.0f / (float)(D_F * D_F);
    if (idx < D_F) { rf[idx] = 1.0f / D_F; cf[idx] = 1.0f / D_F; }
    if (idx == 0) sc[5] = 1.0f;   // pi_f sum
}

// t1[i] = sum_k X[i,k]^2 * r[k] ; (blockIdx.y==1 -> Y,c -> t2)
__global__ void k_t1t2(const float* __restrict__ X, const float* __restrict__ Y,
                       const float* __restrict__ rf, const float* __restrict__ cf,
                       float* __restrict__ t1, float* __restrict__ t2)
{
    __shared__ float w[D_F];
    const float* src; const float* ws; float* dst;
    if (blockIdx.y == 0) { src = X; ws = rf; dst = t1; }
    else                 { src = Y; ws = cf; dst = t2; }
    if (threadIdx.x < D_F) w[threadIdx.x] = ws[threadIdx.x];
    __syncthreads();
    int i = blockIdx.x * 256 + threadIdx.x;
    const float* row = src + (size_t)i * D_F;
    float acc = 0.f;
    for (int k = 0; k < D_F; ++k) { float v = row[k]; acc += v * v * w[k]; }
    dst[i] = acc;
}

// ---------------------------------------------------------------------------
// min/max over 16M elements (two-stage, deterministic)
// ---------------------------------------------------------------------------
__global__ void k_minmax1(const float* __restrict__ C,
                          float* __restrict__ pmin, float* __restrict__ pmax)
{
    __shared__ float smin[256], smax[256];
    int tid = threadIdx.x;
    float lo = 3.4e38f, hi = -3.4e38f;
    int stride = gridDim.x * 256;
    for (int idx = blockIdx.x * 256 + tid; idx < NTOT; idx += stride) {
        float v = C[idx]; lo = fminf(lo, v); hi = fmaxf(hi, v);
    }
    smin[tid] = lo; smax[tid] = hi; __syncthreads();
    for (int off = 128; off > 0; off >>= 1) {
        if (tid < off) {
            smin[tid] = fminf(smin[tid], smin[tid + off]);
            smax[tid] = fmaxf(smax[tid], smax[tid + off]);
        }
        __syncthreads();
    }
    if (tid == 0) { pmin[blockIdx.x] = smin[0]; pmax[blockIdx.x] = smax[0]; }
}

__global__ void k_minmax2(const float* __restrict__ pmin, const float* __restrict__ pmax,
                          float* __restrict__ sc)
{
    __shared__ float smin[256], smax[256];
    int tid = threadIdx.x;
    float lo = 3.4e38f, hi = -3.4e38f;
    for (int i = tid; i < 1024; i += 256) {
        lo = fminf(lo, pmin[i]); hi = fmaxf(hi, pmax[i]);
    }
    smin[tid] = lo; smax[tid] = hi; __syncthreads();
    for (int off = 128; off > 0; off >>= 1) {
        if (tid < off) {
            smin[tid] = fminf(smin[tid], smin[tid + off]);
            smax[tid] = fmaxf(smax[tid], smax[tid + off]);
        }
        __syncthreads();
    }
    if (tid == 0) { sc[0] = smin[0]; sc[1] = smax[0]; }
}

// In-place: K = exp(-C_norm / eps), with degenerate-scale guard (C_norm = 0).
__global__ void k_exp(float* __restrict__ Kb, const float* __restrict__ sc)
{
    float cmin = sc[0], cmax = sc[1];
    float scale = cmax - cmin;
    float inv = (scale > 1e-8f) ? (1.0f / scale) : 0.0f;
    int stride = gridDim.x * 256;
    for (int idx = blockIdx.x * 256 + threadIdx.x; idx < NTOT; idx += stride) {
        float cn = (Kb[idx] - cmin) * inv;
        Kb[idx] = __expf(cn * (-1.0f / EPSI));
    }
}

__global__ void k_initb(float* __restrict__ b)
{
    int i = blockIdx.x * 256 + threadIdx.x;
    if (i < M_S) b[i] = 1.0f;
}

// ---------------------------------------------------------------------------
// Row matvec: u_i = dot(K[i,:], b)
//   MODE 0: out[i] = (mu/(u_i+1e-16))^gamma      (Sinkhorn a-update)
//   MODE 1: out[i] = a_in[i] * u_i               (for pi_s sums)
// One block (256 thr) per row, coalesced.
// ---------------------------------------------------------------------------
template<int MODE>
__global__ void __launch_bounds__(256)
k_rowmv(const float* __restrict__ Kb, const float* __restrict__ b,
        const float* __restrict__ a_in, float* __restrict__ out,
        float mu, float gamma)
{
    __shared__ float red[256];
    const int i = blockIdx.x;
    const float* row = Kb + (size_t)i * M_S;
    float s = 0.f;
    for (int j = threadIdx.x; j < M_S; j += 256) s += row[j] * b[j];
    red[threadIdx.x] = s; __syncthreads();
    for (int off = 128; off > 0; off >>= 1) {
        if (threadIdx.x < off) red[threadIdx.x] += red[threadIdx.x + off];
        __syncthreads();
    }
    if (threadIdx.x == 0) {
        float u = red[0];
        if (MODE == 0) out[i] = __powf(mu / (u + 1e-16f), gamma);
        else           out[i] = a_in[i] * u;
    }
}

// Column matvec: v_j = sum_i K[i,j]*a[i]; b_j = nu/(v_j+1e-16); v saved.
// Block = 32 cols x 8 row-lanes (one row-lane per wave -> coalesced 128B reads).
__global__ void __launch_bounds__(256)
k_colmv(const float* __restrict__ Kb, const float* __restrict__ a,
        float* __restrict__ b, float* __restrict__ vsave, float nu)
{
    __shared__ float part[8][32];
    const int j0 = blockIdx.x * 32;
    const int c  = threadIdx.x & 31;
    const int r  = threadIdx.x >> 5;
    float s = 0.f;
    for (int i = r; i < N_S; i += 8) s += Kb[(size_t)i * M_S + j0 + c] * a[i];
    part[r][c] = s; __syncthreads();
    if (threadIdx.x < 32) {
        float v = 0.f;
        for (int q = 0; q < 8; ++q) v += part[q][threadIdx.x];
        vsave[j0 + threadIdx.x] = v;
        b[j0 + threadIdx.x] = nu / (v + 1e-16f);
    }
}

// S = sum_i rowdot[i];  scale_s = pif_sum/(S+1e-16);  pis_sum = S*scale_s
__global__ void k_reduceS(const float* __restrict__ rowdot, float* __restrict__ sc)
{
    __shared__ float red[256];
    float s = 0.f;
    for (int i = threadIdx.x; i < N_S; i += 256) s += rowdot[i];
    red[threadIdx.x] = s; __syncthreads();
    for (int off = 128; off > 0; off >>= 1) {
        if (threadIdx.x < off) red[threadIdx.x] += red[threadIdx.x + off];
        __syncthreads();
    }
    if (threadIdx.x == 0) {
        float S = red[0];
        float scale = sc[5] / (S + 1e-16f);
        sc[3] = scale;        // scale_s
        sc[4] = S * scale;    // pi_s sum
    }
}

// In-place: pi_s[i,j] = scale_s * a[i] * K[i,j] * b[j]
__global__ void k_scale_pis(float* __restrict__ Kb, const float* __restrict__ a,
                            const float* __restrict__ b, const float* __restrict__ sc)
{
    float scale = sc[3];
    int stride = gridDim.x * 256;
    for (int idx = blockIdx.x * 256 + threadIdx.x; idx < NTOT; idx += stride) {
        int i = idx >> 12;       // / 4096
        int j = idx & 4095;
        Kb[idx] *= scale * a[i] * b[j];
    }
}

// r_s[i] = rowdot[i]*scale_s ; c_s[j] = b[j]*v[j]*scale_s
__global__ void k_rs_cs(const float* __restrict__ rowdot, const float* __restrict__ b,
                        const float* __restrict__ vvec, const float* __restrict__ sc,
                        float* __restrict__ rs, float* __restrict__ cs)
{
    int i = blockIdx.x * 256 + threadIdx.x;
    float scale = sc[3];
    if (i < N_S) {
        rs[i] = rowdot[i] * scale;
        cs[i] = b[i] * vvec[i] * scale;
    }
}

// t1f[k] = sum_i X[i,k]^2 * r_s[i] ; (blockIdx.y==1 -> Y,c_s -> t2f)
__global__ void k_t1t2f(const float* __restrict__ X, const float* __restrict__ Y,
                        const float* __restrict__ rs, const float* __restrict__ cs,
                        float* __restrict__ t1f, float* __restrict__ t2f)
{
    __shared__ float red[256];
    const float* src; const float* w; float* dst;
    if (blockIdx.y == 0) { src = X; w = rs; dst = t1f; }
    else                 { src = Y; w = cs; dst = t2f; }
    const int k = blockIdx.x;
    float acc = 0.f;
    for (int i = threadIdx.x; i < N_S; i += 256) {
        float v = src[(size_t)i * D_F + k];
        acc += v * v * w[i];
    }
    red[threadIdx.x] = acc; __syncthreads();
    for (int off = 128; off > 0; off >>= 1) {
        if (threadIdx.x < off) red[threadIdx.x] += red[threadIdx.x + off];
        __syncthreads();
    }
    if (threadIdx.x == 0) dst[k] = red[0];
}

// ---------------------------------------------------------------------------
// Feature-side Sinkhorn on 128x128 C_f, fully inside one workgroup.
// Transforms Cf -> Kf in place (global ws, L2-hot), 100 iters (gamma=1),
// writes scaled pi_f to d_out, plus its row/col sums + total for next outer.
// ---------------------------------------------------------------------------
__global__ void __launch_bounds__(256)
k_small_sinkhorn(float* __restrict__ Cf, float* __restrict__ pif_out,
                 float* __restrict__ rf, float* __restrict__ cf,
                 float* __restrict__ sc)
{
    __shared__ float red[256];
    __shared__ float a[D_F], b[D_F], u2[D_F], v2[D_F];
    __shared__ float shs[2];
    const int tid = threadIdx.x;
    const float muf = 1.0f / D_F;

    // min / max of Cf
    float lo = 3.4e38f, hi = -3.4e38f;
    for (int e = tid; e < D_F * D_F; e += 256) {
        float v = Cf[e]; lo = fminf(lo, v); hi = fmaxf(hi, v);
    }
    red[tid] = lo; __syncthreads();
    for (int off = 128; off > 0; off >>= 1) {
        if (tid < off) red[tid] = fminf(red[tid], red[tid + off]);
        __syncthreads();
    }
    if (tid == 0) shs[0] = red[0];
    __syncthreads();
    red[tid] = hi; __syncthreads();
    for (int off = 128; off > 0; off >>= 1) {
        if (tid < off) red[tid] = fmaxf(red[tid], red[tid + off]);
        __syncthreads();
    }
    if (tid == 0) shs[1] = red[0];
    __syncthreads();

    float cmin = shs[0];
    float scale = shs[1] - cmin;
    float inv = (scale > 1e-8f) ? (1.0f / scale) : 0.0f;
    for (int e = tid; e < D_F * D_F; e += 256)
        Cf[e] = __expf((Cf[e] - cmin) * inv * (-1.0f / EPSI));
    if (tid < D_F) b[tid] = 1.0f;
    __syncthreads();

    const int ih = tid >> 1;    // row/col index 0..127
    const int hh = tid & 1;     // which half of the 128-sum

    for (int it = 0; it < MAX_INNER; ++it) {
        // a = mu / (K b + eps)   (gamma = 1)
        float s = 0.f;
        {
            const float* rowp = Cf + ih * D_F + hh * 64;
            for (int j = 0; j < 64; ++j) s += rowp[j] * b[hh * 64 + j];
        }
        red[tid] = s; __syncthreads();
        if (tid < D_F) a[tid] = muf / (red[2 * tid] + red[2 * tid + 1] + 1e-16f);
        __syncthreads();
        // b = nu / (K^T a + eps)
        s = 0.f;
        for (int i = 0; i < 64; ++i) s += Cf[(hh * 64 + i) * D_F + ih] * a[hh * 64 + i];
        red[tid] = s; __syncthreads();
        if (tid < D_F) b[tid] = muf / (red[2 * tid] + red[2 * tid + 1] + 1e-16f);
        __syncthreads();
    }

    // u2 = K b (final b), v2 = K^T a, Sf = sum a*u2
    {
        float s = 0.f;
        const float* rowp = Cf + ih * D_F + hh * 64;
        for (int j = 0; j < 64; ++j) s += rowp[j] * b[hh * 64 + j];
        red[tid] = s; __syncthreads();
        if (tid < D_F) u2[tid] = red[2 * tid] + red[2 * tid + 1];
        __syncthreads();
        s = 0.f;
        for (int i = 0; i < 64; ++i) s += Cf[(hh * 64 + i) * D_F + ih] * a[hh * 64 + i];
        red[tid] = s; __syncthreads();
        if (tid < D_F) v2[tid] = red[2 * tid] + red[2 * tid + 1];
        __syncthreads();
        red[tid] = (tid < D_F) ? a[tid] * u2[tid] : 0.f;
        __syncthreads();
        for (int off = 128; off > 0; off >>= 1) {
            if (tid < off) red[tid] += red[tid + off];
            __syncthreads();
        }
        if (tid == 0) {
            float Sf = red[0];
            float sf = sc[4] / (Sf + 1e-16f);   // pi_s.sum / pi_f_new.sum
            shs[0] = sf;
            sc[5] = Sf * sf;                    // new pi_f sum for next outer
        }
        __syncthreads();
    }
    float sf = shs[0];
    for (int e = tid; e < D_F * D_F; e += 256) {
        int i = e >> 7, j = e & 127;
        pif_out[e] = a[i] * Cf[e] * b[j] * sf;
    }
    if (tid < D_F) {
        rf[tid] = a[tid] * u2[tid] * sf;
        cf[tid] = b[tid] * v2[tid] * sf;
    }
}

// ---------------------------------------------------------------------------
extern "C" void kernel_launch(void* const* d_in, const int* in_sizes, int n_in,
                              void* d_out, int out_size, void* d_ws, size_t ws_size,
                              hipStream_t stream)
{
    const float* X = (const float*)d_in[0];
    const float* Y = (const float*)d_in[1];
    float* out = (float*)d_out;
    float* pis = out;                       // [4096*4096]: C_s -> K -> pi_s, in place
    float* pif = out + (size_t)NTOT;        // [128*128]

    float* ws     = (float*)d_ws;           // needs ~4.4 MB total
    float* Amat   = ws;                     // 4096*128  (X @ pi_f)
    float* Z      = Amat + N_S * D_F;       // 4096*128  (pi_s @ Y)
    float* t1     = Z + N_S * D_F;          // 4096
    float* t2     = t1 + N_S;               // 4096
    float* avec   = t2 + M_S;               // 4096
    float* bvec   = avec + N_S;             // 4096
    float* vvec   = bvec + M_S;             // 4096
    float* rowdot = vvec + M_S;             // 4096
    float* rs     = rowdot + N_S;           // 4096
    float* cs     = rs + N_S;               // 4096
    float* pmin   = cs + M_S;               // 1024
    float* pmax   = pmin + 1024;            // 1024
    float* Cf     = pmax + 1024;            // 128*128
    float* t1f    = Cf + D_F * D_F;         // 128
    float* t2f    = t1f + D_F;              // 128
    float* rf     = t2f + D_F;              // 128 (pi_f row sums)
    float* cf     = rf + D_F;               // 128 (pi_f col sums)
    float* sc     = cf + D_F;               // 8 scalars: cmin,cmax,-,scale_s,pis_sum,pif_sum

    const float gamma_s = 5.0f / (5.0f + EPSI);
    const float mu_s = 1.0f / N_S;
    const float nu_s = 1.0f / M_S;

    k_init<<<64, 256, 0, stream>>>(pif, rf, cf, sc);

    for (int outer = 0; outer < MAX_OUTER; ++outer) {
        // ---- sample-side cost C_s = t1 + t2 - 2 * (X pi_f) Y^T ----
        k_t1t2<<<dim3(16, 2), 256, 0, stream>>>(X, Y, rf, cf, t1, t2);
        // Amat = X @ pi_f   [4096x128] = [4096x128][128x128]
        wmma_gemm_f32<0, 0, 0><<<256, 256, 0, stream>>>(X, D_F, pif, D_F, Amat, D_F,
                                                        D_F, D_F, 1.0f, nullptr, nullptr);
        // C_s (into pis buffer) = -2 * Amat @ Y^T + t1[i] + t2[j]
        wmma_gemm_f32<0, 1, 1><<<8192, 256, 0, stream>>>(Amat, D_F, Y, D_F, pis, M_S,
                                                         M_S, D_F, -2.0f, t1, t2);
        // ---- K = exp(-normalized C_s / eps), in place ----
        k_minmax1<<<1024, 256, 0, stream>>>(pis, pmin, pmax);
        k_minmax2<<<1, 256, 0, stream>>>(pmin, pmax, sc);
        k_exp<<<4096, 256, 0, stream>>>(pis, sc);
        // ---- Sinkhorn (100 iters) over L2-resident K ----
        k_initb<<<16, 256, 0, stream>>>(bvec);
        for (int it = 0; it < MAX_INNER; ++it) {
            k_rowmv<0><<<4096, 256, 0, stream>>>(pis, bvec, nullptr, avec, mu_s, gamma_s);
            k_colmv<<<128, 256, 0, stream>>>(pis, avec, bvec, vvec, nu_s);
        }
        // rowdot[i] = a_i * (K b)_i; S; scale; pi_s in place; row/col sums
        k_rowmv<1><<<4096, 256, 0, stream>>>(pis, bvec, avec, rowdot, mu_s, gamma_s);
        k_reduceS<<<1, 256, 0, stream>>>(rowdot, sc);
        k_scale_pis<<<4096, 256, 0, stream>>>(pis, avec, bvec, sc);
        k_rs_cs<<<16, 256, 0, stream>>>(rowdot, bvec, vvec, sc, rs, cs);
        // ---- feature-side cost C_f = t1f + t2f - 2 * X^T pi_s Y ----
        k_t1t2f<<<dim3(128, 2), 256, 0, stream>>>(X, Y, rs, cs, t1f, t2f);
        // Z = pi_s @ Y   [4096x128], K = 4096
        wmma_gemm_f32<0, 0, 0><<<256, 256, 0, stream>>>(pis, M_S, Y, D_F, Z, D_F,
                                                        D_F, M_S, 1.0f, nullptr, nullptr);
        // C_f = -2 * X^T @ Z + t1f + t2f   [128x128], K = 4096
        wmma_gemm_f32<1, 0, 1><<<8, 256, 0, stream>>>(X, D_F, Z, D_F, Cf, D_F,
                                                      D_F, M_S, -2.0f, t1f, t2f);
        // ---- 128x128 Sinkhorn + pi_f write + sums, one workgroup ----
        k_small_sinkhorn<<<1, 256, 0, stream>>>(Cf, pif, rf, cf, sc);
    }
}